// GAT_57458072485947
// MI455X (gfx1250) — compile-verified
//
#include <hip/hip_runtime.h>
#include <math.h>

typedef __attribute__((ext_vector_type(2))) float v2f;
typedef __attribute__((ext_vector_type(8))) float v8f;
typedef __attribute__((ext_vector_type(4))) unsigned int u32x4;
typedef __attribute__((ext_vector_type(8))) int i32x8;
typedef __attribute__((ext_vector_type(4))) int i32x4;

#define NEG_SLOPE 0.2f

// -------- atomic float max via signed/unsigned bit trick --------
// Works with init bits 0xFFFFFFFF (NaN bits act as -inf under this ordering).
__device__ __forceinline__ void atomicMaxF(float* addr, float v) {
    if (v >= 0.0f) {
        atomicMax((int*)addr, __float_as_int(v));
    } else {
        atomicMin((unsigned int*)addr, __float_as_uint(v));
    }
}

__device__ __forceinline__ float lrelu(float x) {
    return x > 0.0f ? x : NEG_SLOPE * x;
}

// ================= Layer-1 GEMM: xw = x[N,256] @ W1[256,128] =================
// Block = 8 waves; W1 staged through LDS in 32-row K-chunks (16 KB) shared by
// all 8 waves (8x cut in L2 reads). One wave per 16-row stripe computes all
// 8 column tiles with exact-fp32 V_WMMA_F32_16X16X4_F32.
__global__ __launch_bounds__(256) void gemm1_wmma(
    const float* __restrict__ x, const float* __restrict__ W1,
    float* __restrict__ xw, int N) {
    __shared__ float sW[32 * 128];   // 16 KB K-chunk of W1
    const int lane = threadIdx.x & 31;
    const int wave = threadIdx.x >> 5;
    const int row0 = (blockIdx.x * 8 + wave) * 16;
    const int half = lane >> 4;      // K sub-pair selector for A/B frags
    const int lm   = lane & 15;      // M (A) / N (B,C) index

    int arow = row0 + lm;
    if (arow >= N) arow = N - 1;
    const float* xrow = x + (size_t)arow * 256;

    v8f acc[8] = {};
    for (int kc = 0; kc < 256; kc += 32) {
        __syncthreads();
        #pragma unroll
        for (int i = 0; i < 4; ++i) {
            int idx = threadIdx.x + i * 256;   // 1024 float4s = 4096 floats
            ((float4*)sW)[idx] = ((const float4*)(W1 + (size_t)kc * 128))[idx];
        }
        __syncthreads();
        if (row0 < N) {                        // wave-uniform: EXEC stays all-1s
            #pragma unroll
            for (int k0 = 0; k0 < 32; k0 += 4) {
                const int kk = k0 + 2 * half;
                v2f a;
                a.x = xrow[kc + kk];
                a.y = xrow[kc + kk + 1];
                const float* w0 = sW + kk * 128 + lm;
                #pragma unroll
                for (int t = 0; t < 8; ++t) {
                    v2f b;
                    b.x = w0[t * 16];
                    b.y = w0[t * 16 + 128];
                    acc[t] = __builtin_amdgcn_wmma_f32_16x16x4_f32(
                        false, a, false, b, (short)0, acc[t], false, false);
                }
            }
        }
    }
    if (row0 < N) {
        #pragma unroll
        for (int t = 0; t < 8; ++t) {
            #pragma unroll
            for (int r = 0; r < 8; ++r) {
                const int row = row0 + r + 8 * half;
                if (row < N) xw[(size_t)row * 128 + t * 16 + lm] = acc[t][r];
            }
        }
    }
}

// ================= Layer-2 GEMM: hw = h[N,128] @ W2p[128,48] =================
// W2p (24 KB) pulled into LDS once per block via the Tensor Data Mover
// (tensor_load_to_lds + s_wait_tensorcnt); cooperative-load fallback otherwise.
__global__ __launch_bounds__(256) void gemm2_wmma(
    const float* __restrict__ h, const float* __restrict__ W2p,
    float* __restrict__ hw, int N) {
    __shared__ float sW[128 * 48];   // 24 KB
#if defined(__gfx1250__) && __has_builtin(__builtin_amdgcn_tensor_load_to_lds)
    if ((threadIdx.x >> 5) == 0) {   // one wave issues the TDM op (EXEC ignored)
        unsigned long long ga = (unsigned long long)(uintptr_t)W2p;
        u32x4 g0;
        g0[0] = 1u;                                  // count=1, user descriptor
        g0[1] = 0u;                                  // lds_addr = 0 (sW at base)
        g0[2] = (unsigned int)(ga & 0xFFFFFFFFull);  // global_addr[31:0]
        g0[3] = (unsigned int)((ga >> 32) & 0x1FFFFFFull) | 0x80000000u; // addr[56:32] | type=2
        i32x8 g1;
        g1[0] = 0x00020000;          // workgroup_mask=0, data_size=2 (4 bytes)
        g1[1] = (48 << 16);          // tensor_dim0 = 48 (lo16)
        g1[2] = (128 << 16);         // tensor_dim0 hi16=0, tensor_dim1 = 128 (lo16)
        g1[3] = (48 << 16);          // tensor_dim1 hi16=0, tile_dim0 = 48
        g1[4] = 128;                 // tile_dim1 = 128, tile_dim2 = 0
        g1[5] = 48;                  // tensor_dim0_stride = 48
        g1[6] = 0;                   // stride hi, tensor_dim1_stride lo (unused)
        g1[7] = 0;
        i32x4 z4 = {0, 0, 0, 0};
        i32x8 z8 = {0, 0, 0, 0, 0, 0, 0, 0};
        __builtin_amdgcn_tensor_load_to_lds(g0, g1, z4, z4, z8, 0);
        __builtin_amdgcn_s_wait_tensorcnt(0);
    }
#else
    for (int i = threadIdx.x; i < 128 * 48 / 4; i += 256)
        ((float4*)sW)[i] = ((const float4*)W2p)[i];
#endif
    __syncthreads();

    const int lane = threadIdx.x & 31;
    const int wave = threadIdx.x >> 5;
    const int row0 = (blockIdx.x * 8 + wave) * 16;
    if (row0 >= N) return;           // wave-uniform exit (after barrier)
    const int half = lane >> 4;
    const int lm   = lane & 15;

    int arow = row0 + lm;
    if (arow >= N) arow = N - 1;
    const float* hrow = h + (size_t)arow * 128;

    v8f acc[3] = {};
    for (int k0 = 0; k0 < 128; k0 += 4) {
        const int kk = k0 + 2 * half;
        v2f a;
        a.x = hrow[kk];
        a.y = hrow[kk + 1];
        const float* w0 = sW + kk * 48 + lm;
        #pragma unroll
        for (int t = 0; t < 3; ++t) {
            v2f b;
            b.x = w0[t * 16];
            b.y = w0[t * 16 + 48];
            acc[t] = __builtin_amdgcn_wmma_f32_16x16x4_f32(
                false, a, false, b, (short)0, acc[t], false, false);
        }
    }
    #pragma unroll
    for (int t = 0; t < 3; ++t) {
        #pragma unroll
        for (int r = 0; r < 8; ++r) {
            const int row = row0 + r + 8 * half;
            if (row < N) hw[(size_t)row * 48 + t * 16 + lm] = acc[t][r];
        }
    }
}

// ---- pad W2[128,40] -> W2p[128,48] (cols 40..47 zero) ----
__global__ void pack_w2(const float* __restrict__ W2, float* __restrict__ W2p) {
    int i = blockIdx.x * blockDim.x + threadIdx.x;
    if (i >= 128 * 48) return;
    int r = i / 48, c = i % 48;
    W2p[i] = (c < 40) ? W2[r * 40 + c] : 0.0f;
}

// ---- per-(node,head) attention scores: a = sum_c xw[n,h,c]*att[h,c] ----
__global__ void attn_scores1(const float* __restrict__ xw,
                             const float* __restrict__ att_src,
                             const float* __restrict__ att_dst,
                             float* __restrict__ as_, float* __restrict__ ad_,
                             int N) {
    int i = blockIdx.x * blockDim.x + threadIdx.x;
    if (i >= N * 8) return;
    int n = i >> 3, hd = i & 7;
    const float* v = xw + (size_t)n * 128 + hd * 16;
    float s = 0.0f, d = 0.0f;
    #pragma unroll
    for (int c = 0; c < 16; ++c) {
        s += v[c] * att_src[hd * 16 + c];
        d += v[c] * att_dst[hd * 16 + c];
    }
    as_[i] = s; ad_[i] = d;
}

__global__ void attn_scores2(const float* __restrict__ hw,
                             const float* __restrict__ att_src,
                             const float* __restrict__ att_dst,
                             float* __restrict__ as_, float* __restrict__ ad_,
                             int N) {
    int n = blockIdx.x * blockDim.x + threadIdx.x;
    if (n >= N) return;
    const float* v = hw + (size_t)n * 48;
    float s = 0.0f, d = 0.0f;
    #pragma unroll
    for (int c = 0; c < 40; ++c) {
        s += v[c] * att_src[c];
        d += v[c] * att_dst[c];
    }
    as_[n] = s; ad_[n] = d;
}

// ---- edge logits + segment max (layer 1, H=8) ----
__global__ void edge_logit_max1(const int* __restrict__ ei, int E,
                                const float* __restrict__ as_,
                                const float* __restrict__ ad_,
                                float* __restrict__ logit, float* __restrict__ m) {
    int e = blockIdx.x * blockDim.x + threadIdx.x;
    if (e >= E) return;
    int src = ei[e], dst = ei[E + e];
    #pragma unroll
    for (int hd = 0; hd < 8; ++hd) {
        float l = lrelu(as_[src * 8 + hd] + ad_[dst * 8 + hd]);
        logit[(size_t)e * 8 + hd] = l;
        atomicMaxF(&m[dst * 8 + hd], l);
    }
}

__global__ void edge_logit_max2(const int* __restrict__ ei, int E,
                                const float* __restrict__ as_,
                                const float* __restrict__ ad_,
                                float* __restrict__ logit, float* __restrict__ m) {
    int e = blockIdx.x * blockDim.x + threadIdx.x;
    if (e >= E) return;
    int src = ei[e], dst = ei[E + e];
    float l = lrelu(as_[src] + ad_[dst]);
    logit[e] = l;
    atomicMaxF(&m[dst], l);
}

// ---- exp(logit - m) in place + segment sum ----
__global__ void edge_exp_sum1(const int* __restrict__ ei, int E,
                              const float* __restrict__ m,
                              float* __restrict__ logit, float* __restrict__ s) {
    int e = blockIdx.x * blockDim.x + threadIdx.x;
    if (e >= E) return;
    int dst = ei[E + e];
    #pragma unroll
    for (int hd = 0; hd < 8; ++hd) {
        float mv = m[dst * 8 + hd];
        mv = (mv == mv && fabsf(mv) <= 3.0e38f) ? mv : 0.0f;  // isfinite -> else 0
        float ee = __expf(logit[(size_t)e * 8 + hd] - mv);
        logit[(size_t)e * 8 + hd] = ee;
        atomicAdd(&s[dst * 8 + hd], ee);
    }
}

__global__ void edge_exp_sum2(const int* __restrict__ ei, int E,
                              const float* __restrict__ m,
                              float* __restrict__ logit, float* __restrict__ s) {
    int e = blockIdx.x * blockDim.x + threadIdx.x;
    if (e >= E) return;
    int dst = ei[E + e];
    float mv = m[dst];
    mv = (mv == mv && fabsf(mv) <= 3.0e38f) ? mv : 0.0f;
    float ee = __expf(logit[e] - mv);
    logit[e] = ee;
    atomicAdd(&s[dst], ee);
}

// ---- weighted scatter: out[dst] += alpha * xw[src]  (layer 1: 128 ch) ----
// One wave32 per edge: edge scalars become wave-uniform (scalar loads);
// each lane moves 4 channels with one b128 load + 4 fp32 atomics.
__global__ __launch_bounds__(256) void aggregate1(
    const int* __restrict__ ei, int E,
    const float* __restrict__ ew,
    const float* __restrict__ enum_,   // exp values [E,8]
    const float* __restrict__ s,
    const float* __restrict__ xw,
    float* __restrict__ out) {
    const int wave = threadIdx.x >> 5;
    const int lane = threadIdx.x & 31;
    const int e = blockIdx.x * 8 + wave;
    if (e >= E) return;
    const int src = ei[e];          // wave-uniform -> scalar load
    const int dst = ei[E + e];
    const float w = ew[e];
    const int hd = lane >> 2;       // 4 channels per lane, head = (lane*4)/16
    const float alpha =
        enum_[(size_t)e * 8 + hd] / (s[dst * 8 + hd] + 1e-16f) * w;
    const float4 xv = *(const float4*)(xw + (size_t)src * 128 + lane * 4);
    float* o = out + (size_t)dst * 128 + lane * 4;
    atomicAdd(o + 0, alpha * xv.x);
    atomicAdd(o + 1, alpha * xv.y);
    atomicAdd(o + 2, alpha * xv.z);
    atomicAdd(o + 3, alpha * xv.w);
}

// ---- layer 2: 40 channels, hw leading dim 48; one wave32 per edge ----
__global__ __launch_bounds__(256) void aggregate2(
    const int* __restrict__ ei, int E,
    const float* __restrict__ ew,
    const float* __restrict__ enum_,   // [E]
    const float* __restrict__ s,
    const float* __restrict__ hw,
    float* __restrict__ out) {
    const int wave = threadIdx.x >> 5;
    const int lane = threadIdx.x & 31;
    const int e = blockIdx.x * 8 + wave;
    if (e >= E) return;
    const int src = ei[e];
    const int dst = ei[E + e];
    const float alpha = enum_[e] / (s[dst] + 1e-16f) * ew[e];
    const float* hv = hw + (size_t)src * 48;
    float* o = out + (size_t)dst * 40;
    atomicAdd(o + lane, alpha * hv[lane]);
    if (lane < 8) atomicAdd(o + 32 + lane, alpha * hv[32 + lane]);
}

// ---- h = elu(out1 + b1), in place ----
__global__ void elu_bias(float* __restrict__ io, const float* __restrict__ b1, int N) {
    long long i = (long long)blockIdx.x * blockDim.x + threadIdx.x;
    if (i >= (long long)N * 128) return;
    float v = io[i] + b1[(int)(i & 127)];
    io[i] = v > 0.0f ? v : expm1f(v);
}

// ---- final: log_softmax(out2 + b2) over 40 classes ----
__global__ void bias_logsoftmax(const float* __restrict__ out2,
                                const float* __restrict__ b2,
                                float* __restrict__ y, int N) {
    int n = blockIdx.x * blockDim.x + threadIdx.x;
    if (n >= N) return;
    float v[40];
    float mx = -3.4e38f;
    #pragma unroll
    for (int c = 0; c < 40; ++c) {
        v[c] = out2[(size_t)n * 40 + c] + b2[c];
        mx = fmaxf(mx, v[c]);
    }
    float sum = 0.0f;
    #pragma unroll
    for (int c = 0; c < 40; ++c) sum += __expf(v[c] - mx);
    float lse = mx + logf(sum);
    #pragma unroll
    for (int c = 0; c < 40; ++c) y[(size_t)n * 40 + c] = v[c] - lse;
}

extern "C" void kernel_launch(void* const* d_in, const int* in_sizes, int n_in,
                              void* d_out, int out_size, void* d_ws, size_t ws_size,
                              hipStream_t stream) {
    const float* x    = (const float*)d_in[0];
    const int*   ei   = (const int*)d_in[1];
    const float* ew   = (const float*)d_in[2];
    const float* W1   = (const float*)d_in[3];
    const float* as1  = (const float*)d_in[4];
    const float* ad1  = (const float*)d_in[5];
    const float* b1   = (const float*)d_in[6];
    const float* W2   = (const float*)d_in[7];
    const float* as2  = (const float*)d_in[8];
    const float* ad2  = (const float*)d_in[9];
    const float* b2   = (const float*)d_in[10];
    float* y = (float*)d_out;

    const int N = in_sizes[0] / 256;
    const int E = in_sizes[2];

    // ---- workspace layout (floats) ----
    float* p = (float*)d_ws;
    float* xw     = p; p += (size_t)N * 128;
    float* a_s1   = p; p += (size_t)N * 8;
    float* a_d1   = p; p += (size_t)N * 8;
    float* m1     = p; p += (size_t)N * 8;
    float* s1     = p; p += (size_t)N * 8;
    float* logit1 = p; p += (size_t)E * 8;
    float* out1   = p; p += (size_t)N * 128;   // becomes h (in-place elu)
    float* w2p    = p; p += (size_t)128 * 48;
    float* hw     = p; p += (size_t)N * 48;
    float* a_s2   = p; p += (size_t)N;
    float* a_d2   = p; p += (size_t)N;
    float* m2     = p; p += (size_t)N;
    float* s2     = p; p += (size_t)N;
    float* logit2 = p; p += (size_t)E;
    float* out2   = p; p += (size_t)N * 40;

    // ---- init (graph-capturable memsets) ----
    (void)hipMemsetAsync(out1, 0,    (size_t)N * 128 * sizeof(float), stream);
    (void)hipMemsetAsync(s1,   0,    (size_t)N * 8   * sizeof(float), stream);
    (void)hipMemsetAsync(m1,   0xFF, (size_t)N * 8   * sizeof(float), stream); // NaN == -inf under bit-trick
    (void)hipMemsetAsync(out2, 0,    (size_t)N * 40  * sizeof(float), stream);
    (void)hipMemsetAsync(s2,   0,    (size_t)N       * sizeof(float), stream);
    (void)hipMemsetAsync(m2,   0xFF, (size_t)N       * sizeof(float), stream);

    const int rowTiles = (N + 15) / 16;
    const int gemmBlocks = (rowTiles + 7) / 8;
    const int edgeWaveBlocks = (E + 7) / 8;

    // ================= Layer 1 =================
    gemm1_wmma<<<gemmBlocks, 256, 0, stream>>>(x, W1, xw, N);
    attn_scores1<<<(N * 8 + 255) / 256, 256, 0, stream>>>(xw, as1, ad1, a_s1, a_d1, N);
    edge_logit_max1<<<(E + 255) / 256, 256, 0, stream>>>(ei, E, a_s1, a_d1, logit1, m1);
    edge_exp_sum1<<<(E + 255) / 256, 256, 0, stream>>>(ei, E, m1, logit1, s1);
    aggregate1<<<edgeWaveBlocks, 256, 0, stream>>>(ei, E, ew, logit1, s1, xw, out1);
    {
        long long total = (long long)N * 128;
        int blocks = (int)((total + 255) / 256);
        elu_bias<<<blocks, 256, 0, stream>>>(out1, b1, N);
    }

    // ================= Layer 2 =================
    pack_w2<<<(128 * 48 + 255) / 256, 256, 0, stream>>>(W2, w2p);
    gemm2_wmma<<<gemmBlocks, 256, 0, stream>>>(out1, w2p, hw, N);
    attn_scores2<<<(N + 255) / 256, 256, 0, stream>>>(hw, as2, ad2, a_s2, a_d2, N);
    edge_logit_max2<<<(E + 255) / 256, 256, 0, stream>>>(ei, E, a_s2, a_d2, logit2, m2);
    edge_exp_sum2<<<(E + 255) / 256, 256, 0, stream>>>(ei, E, m2, logit2, s2);
    aggregate2<<<edgeWaveBlocks, 256, 0, stream>>>(ei, E, ew, logit2, s2, hw, out2);
    bias_logsoftmax<<<(N + 255) / 256, 256, 0, stream>>>(out2, b2, y, N);
}